// SparseGCM_38826504356582
// MI455X (gfx1250) — compile-verified
//
#include <hip/hip_runtime.h>
#include <hip/hip_bf16.h>

// Problem constants (match reference)
#define BB   32
#define NN   2048
#define TAU  16
#define FF   128
#define EE   32768          // 2^15 edges per batch

typedef __attribute__((ext_vector_type(2))) float v2f;
typedef __attribute__((ext_vector_type(8))) float v8f;

// ---------------------------------------------------------------------------
// Kernel 1: zero the per-(b, j) feature accumulator acc[B, TAU, F] in d_ws.
// ---------------------------------------------------------------------------
__global__ void sgcm_zero_acc(float* __restrict__ acc) {
    int i = blockIdx.x * blockDim.x + threadIdx.x;   // 0 .. B*TAU*F-1
    acc[i] = 0.0f;
}

// ---------------------------------------------------------------------------
// Kernel 2: edge scatter-add of RAW features (W_nbr folded out by linearity).
// One wave (32 lanes) per edge; only edges whose dst lands in the window
// [T[b], T[b]+TAU) contribute to the output, all others are skipped.
// The effective node value is x (freshly scattered) if src is in-window,
// else the original node table entry.
// ---------------------------------------------------------------------------
__global__ void __launch_bounds__(256)
sgcm_edge_accum(const float* __restrict__ x,
                const float* __restrict__ nodes,
                const int*   __restrict__ edges,
                const int*   __restrict__ T,
                float*       __restrict__ acc) {
    const int lane  = threadIdx.x & 31;
    const int eg    = blockIdx.x * 8 + (threadIdx.x >> 5);  // global edge id, 0..B*E-1
    const int b     = eg >> 15;                              // E = 2^15
    const int e     = eg & (EE - 1);

    const int* eb = edges + (size_t)b * 2 * EE;
    const int src = eb[e];
    const int dst = eb[EE + e];
    if (src < 0 || dst < 0) return;                          // mask

    const int t0 = T[b];
    const int j  = dst - t0;
    if (j < 0 || j >= TAU) return;                           // dst outside read-back window

    const int   sj   = src - t0;
    const float* srow = (sj >= 0 && sj < TAU)
        ? (x     + ((size_t)b * TAU + sj) * FF)              // overlaid fresh observation
        : (nodes + ((size_t)b * NN  + src) * FF);            // original memory row
    float* arow = acc + ((size_t)b * TAU + j) * FF;

#pragma unroll
    for (int q = 0; q < 4; ++q) {
        const int f = lane + q * 32;                         // coalesced across the wave
        atomicAdd(arow + f, srow[f]);
    }
}

// ---------------------------------------------------------------------------
// Kernel 3: fused dual GEMM + bias + tanh on V_WMMA_F32_16X16X4_F32.
//   out[r, n] = tanh( x[r,:] @ W_self[:,n] + acc[r,:] @ W_nbr[:,n] + bias[n] )
// Rows r = 0..511 (= B*TAU), cols n = 0..127.  One wave32 per 16x16 tile.
// A 16x4 f32 layout: lane L -> row m=L&15, k = 2*(L>>4)+{0,1} (2 VGPRs).
// B 4x16 f32 layout: lane L -> col n=L&15, k = 2*(L>>4)+{0,1} (2 VGPRs).
// C/D 16x16 f32:     VGPR i -> row m = i + 8*(L>>4), col n = L&15.
// ---------------------------------------------------------------------------
__global__ void __launch_bounds__(256)
sgcm_wmma_tanh(const float* __restrict__ x,
               const float* __restrict__ acc,
               const float* __restrict__ Wself,
               const float* __restrict__ Wnbr,
               const float* __restrict__ bias,
               float*       __restrict__ out) {
    const int lane  = threadIdx.x & 31;
    const int tile  = blockIdx.x * 8 + (threadIdx.x >> 5);   // 0..255
    const int tileN = tile & 7;                              // 8 col tiles (F=128)
    const int tileM = tile >> 3;                             // 32 row tiles (512 rows)
    const int p     = lane & 15;
    const int half  = lane >> 4;                             // 0 or 1
    const int row   = tileM * 16 + p;                        // A-row for this lane
    const int col   = tileN * 16 + p;                        // B/D-col for this lane

    v8f c = {};
#pragma unroll 4
    for (int kk = 0; kk < FF; kk += 4) {
        const int ka = kk + half * 2;
        v2f a, w;
        // self term: x @ W_self
        a.x = x[row * FF + ka];
        a.y = x[row * FF + ka + 1];
        w.x = Wself[ka * FF + col];
        w.y = Wself[(ka + 1) * FF + col];
        c = __builtin_amdgcn_wmma_f32_16x16x4_f32(false, a, false, w,
                                                  (short)0, c, false, false);
        // neighbor term: acc @ W_nbr (W_nbr applied post-aggregation)
        a.x = acc[row * FF + ka];
        a.y = acc[row * FF + ka + 1];
        w.x = Wnbr[ka * FF + col];
        w.y = Wnbr[(ka + 1) * FF + col];
        c = __builtin_amdgcn_wmma_f32_16x16x4_f32(false, a, false, w,
                                                  (short)0, c, false, false);
    }

    const float bv = bias[col];
#pragma unroll
    for (int i = 0; i < 8; ++i) {
        const int m = tileM * 16 + i + half * 8;             // C/D VGPR row mapping
        out[m * FF + col] = tanhf(c[i] + bv);
    }
}

// ---------------------------------------------------------------------------
// Launch: inputs in setup_inputs() order:
//   0:x [B,TAU,F] f32   1:nodes [B,N,F] f32   2:edges [B,2,E] i32
//   3:weights (unused)  4:T [B] i32           5:W_self [F,F] f32
//   6:W_nbr [F,F] f32   7:bias [F] f32
// Output: mx [B,TAU,F] f32.
// ---------------------------------------------------------------------------
extern "C" void kernel_launch(void* const* d_in, const int* in_sizes, int n_in,
                              void* d_out, int out_size, void* d_ws, size_t ws_size,
                              hipStream_t stream) {
    const float* x     = (const float*)d_in[0];
    const float* nodes = (const float*)d_in[1];
    const int*   edges = (const int*)  d_in[2];
    const int*   T     = (const int*)  d_in[4];
    const float* Wself = (const float*)d_in[5];
    const float* Wnbr  = (const float*)d_in[6];
    const float* bias  = (const float*)d_in[7];
    float*       out   = (float*)d_out;

    float* acc = (float*)d_ws;                    // B*TAU*F floats = 256 KB

    // 1) zero accumulator (ws is poisoned by harness)
    {
        const int n = BB * TAU * FF;              // 65536
        sgcm_zero_acc<<<n / 256, 256, 0, stream>>>(acc);
    }
    // 2) windowed edge scatter-add (raw features)
    {
        const int nEdges = BB * EE;               // 1M
        sgcm_edge_accum<<<nEdges / 8, 256, 0, stream>>>(x, nodes, edges, T, acc);
    }
    // 3) fused WMMA dual-GEMM + bias + tanh
    {
        // 256 tiles / 8 waves per block = 32 blocks
        sgcm_wmma_tanh<<<32, 256, 0, stream>>>(x, acc, Wself, Wnbr, bias, out);
    }
}